// TagLoss_4836133175850
// MI455X (gfx1250) — compile-verified
//
#include <hip/hip_runtime.h>

typedef float v2f __attribute__((ext_vector_type(2)));
typedef float v8f __attribute__((ext_vector_type(8)));

namespace {
constexpr int B = 32;
constexpr int D = 32;
constexpr int H = 256;
constexpr int W = 256;
constexpr int K = 512;
constexpr int TILE = 16;
constexpr int TILES_PER_DIM = K / TILE;              // 32
constexpr int TILES_PER_IMG = TILES_PER_DIM * TILES_PER_DIM; // 1024
constexpr float SCALE = 1.0f / (float)((long long)K * K * B); // mean over K*K and B
}

// Kernel 1: gather pred[b][d][k] = ebd[b,d,h_k,w_k], and sq[b][k] = mean_d pred^2.
// One thread per (b,k). Writes over k are coalesced; this makes the WMMA kernel's
// reads (lane-major over k) fully coalesced, which matters more.
__global__ void taglbss_gather(const float* __restrict__ ebd,
                               const int* __restrict__ kpts,
                               float* __restrict__ pred,
                               float* __restrict__ sq) {
  int idx = blockIdx.x * blockDim.x + threadIdx.x; // b*K + k
  if (idx >= B * K) return;
  int b = idx >> 9;        // / K
  int k = idx & (K - 1);
  int h = kpts[idx * 2 + 0];
  int w = kpts[idx * 2 + 1];
  const float* src = ebd + ((size_t)b * D) * (size_t)(H * W) + (size_t)h * W + w;
  float* dst = pred + ((size_t)b * D) * K + k;
  float ssum = 0.0f;
#pragma unroll 4
  for (int d = 0; d < D; ++d) {
    float v = src[(size_t)d * (H * W)];
    dst[(size_t)d * K] = v;
    ssum += v * v;
  }
  sq[idx] = ssum * (1.0f / (float)D);
}

// Kernel 2: one wave32 per 16x16 tile of the 512x512 Gram matrix of one image.
// gram_tile = A^T * A with contraction over D=32, done as 8x V_WMMA_F32_16X16X4_F32
// (exact fp32, matching the fp32 reference). Then the elementwise tail
// (expo -> 2/(1+exp) -> squared error vs tag equality) and a wave reduction.
__global__ void taglbss_gram_loss(const float* __restrict__ pred,
                                  const float* __restrict__ sq,
                                  const int* __restrict__ tags,
                                  float* __restrict__ out) {
  const int lane = threadIdx.x & 31;
  const int waveInBlock = threadIdx.x >> 5;
  const int tileId = blockIdx.x * (blockDim.x >> 5) + waveInBlock;

  const int b = tileId / TILES_PER_IMG;
  const int t = tileId - b * TILES_PER_IMG;
  const int krow0 = (t / TILES_PER_DIM) * TILE;
  const int kcol0 = (t - (t / TILES_PER_DIM) * TILES_PER_DIM) * TILE;

  const float* pb = pred + (size_t)b * D * K;

  const int m = lane & 15;     // row/col within tile for this lane's fragment
  const int half = lane >> 4;  // 0: K=0,1 ; 1: K=2,3  (fp32 16x4 A layout)

  v8f acc = {};
#pragma unroll
  for (int d0 = 0; d0 < D; d0 += 4) {
    const int da = d0 + 2 * half;
    v2f a, bfrag;
    // A: 16x4 (M=k-row, K=d). Lane m holds A[m][da], A[m][da+1].
    a.x = pb[(size_t)da * K + krow0 + m];
    a.y = pb[(size_t)(da + 1) * K + krow0 + m];
    // B: 4x16 (K=d, N=k-col). Mirrored striping.
    bfrag.x = pb[(size_t)da * K + kcol0 + m];
    bfrag.y = pb[(size_t)(da + 1) * K + kcol0 + m];
    acc = __builtin_amdgcn_wmma_f32_16x16x4_f32(
        /*neg_a=*/false, a, /*neg_b=*/false, bfrag,
        /*c_mod=*/(short)0, acc, /*reuse_a=*/false, /*reuse_b=*/false);
  }

  // Column-side (N = m) scalars for this lane.
  const float sq_col = sq[b * K + kcol0 + m];
  const int tag_col = tags[b * K + kcol0 + m];

  // C/D f32 16x16 layout: VGPR r, lanes 0-15 -> M=r, lanes 16-31 -> M=r+8; N=lane&15.
  float lsum = 0.0f;
#pragma unroll
  for (int r = 0; r < 8; ++r) {
    const int row = krow0 + r + 8 * half;
    const float gm = acc[r] * (1.0f / (float)D);
    const float sq_row = sq[b * K + row];
    const int tag_row = tags[b * K + row];
    const float expo = sq_row + sq_col - 2.0f * gm;
    const float ps = 2.0f / (1.0f + __expf(expo));
    const float ts = (tag_row == tag_col) ? 1.0f : 0.0f;
    const float diff = ts - ps;
    lsum += diff * diff;
  }

  // Wave32 reduction.
#pragma unroll
  for (int off = 16; off > 0; off >>= 1)
    lsum += __shfl_down(lsum, off, 32);

  if (lane == 0)
    atomicAdd(out, lsum * SCALE);
}

extern "C" void kernel_launch(void* const* d_in, const int* in_sizes, int n_in,
                              void* d_out, int out_size, void* d_ws, size_t ws_size,
                              hipStream_t stream) {
  const float* ebd = (const float*)d_in[0];  // (B, D, H, W) f32
  const int* kpts = (const int*)d_in[1];     // (B, K, 2)
  const int* tags = (const int*)d_in[2];     // (B, K)
  float* out = (float*)d_out;                // scalar

  float* pred = (float*)d_ws;                          // B*D*K floats (2 MB)
  float* sq = pred + (size_t)B * D * K;                // B*K floats (64 KB)

  hipMemsetAsync(out, 0, sizeof(float), stream);

  {
    const int total = B * K;                 // 16384 threads
    const int blk = 256;
    taglbss_gather<<<(total + blk - 1) / blk, blk, 0, stream>>>(ebd, kpts, pred, sq);
  }
  {
    const int totalWaves = B * TILES_PER_IMG; // 32768 waves, one per 16x16 tile
    const int wavesPerBlock = 8;              // 256 threads
    taglbss_gram_loss<<<totalWaves / wavesPerBlock, wavesPerBlock * 32, 0, stream>>>(
        pred, sq, tags, out);
  }
}